// PSL_5866925326343
// MI455X (gfx1250) — compile-verified
//
#include <hip/hip_runtime.h>

// ---------------------------------------------------------------------------
// Problem constants (from reference): B=64, E=256, S=32, H=16, hd=16,
// P_PAIRS=512, K_CLASSES=46, D_FINAL=10.
//
// Math collapse used here (exact): scores[s,h,l,m] = G[s,h,i_l,j_m], and the
// softmax over the 512 m's equals a c_b-weighted softmax over 64 distinct j
// values (c_b = multiplicity of b in j_idx).  Hence o / h / logits depend on
// pairs only through i_idx, so we compute everything on 64 distinct samples.
//
// Scratch (d_ws) layout, floats (needs ~8.4 MB):
//   [0      ) int   i_idx[512]
//   [512    ) int   j_idx[512]
//   [1024   ) int   y[512]
//   [1536   ) int   valid[512]
//   [2048   ) float c_b[64]
//   [2112   ) float cls_cnt[46]
//   [4096   ) float Q[32][16][64][16]   (s,h,a,d)  2 MB
//   [+524288) float K[32][16][64][16]
//   [+524288) float V[32][16][64][16]
//   [+524288) float O[64][32][256]      (a,s,e')
// ---------------------------------------------------------------------------

typedef __attribute__((ext_vector_type(2))) float v2f;
typedef __attribute__((ext_vector_type(8))) float v8f;

__device__ __forceinline__ v8f vz8() {
  v8f z = {0.f, 0.f, 0.f, 0.f, 0.f, 0.f, 0.f, 0.f};
  return z;
}

// fp32 WMMA, D = A(16x4) * B(4x16) + C(16x16).
// A: lane L<16 -> row M=L, {v0,v1}={K0,K1}; lane>=16 -> {K2,K3}.
// B: lane L<16 -> col N=L, {v0,v1}={K0,K1}; lane>=16 -> {K2,K3}.
// C/D: vgpr r -> row r (lanes 0-15) / row r+8 (lanes 16-31), col = lane&15.
__device__ __forceinline__ v8f wmma4(v2f a, v2f b, v8f c) {
  return __builtin_amdgcn_wmma_f32_16x16x4_f32(false, a, false, b,
                                               (short)0, c, false, false);
}

// ---------------------------------------------------------------------------
// Kernel 1: pair enumeration (row-major nonzero with fill), pair class ids,
// key multiplicities c_b, and class counts.  1 block x 64 threads.
// ---------------------------------------------------------------------------
__global__ void __launch_bounds__(64, 1)
PSL_pairs_kernel(const int* __restrict__ labels0,
                 const int* __restrict__ labels1,
                 int* __restrict__ g_i, int* __restrict__ g_j,
                 int* __restrict__ g_y, int* __restrict__ g_valid,
                 float* __restrict__ g_cb, float* __restrict__ g_cls) {
  __shared__ int cnt[64], pre[65];
  __shared__ int si[512], sj[512];
  __shared__ int cb_s[64];
  __shared__ float cc_s[46];
  const int t = threadIdx.x;  // 64 threads
  const int l1 = labels1[t];
  int c = 0;
  for (int j = 0; j < 64; ++j) c += (labels1[j] == l1) ? 1 : 0;
  cnt[t] = c;
  cb_s[t] = 0;
  if (t < 46) cc_s[t] = 0.f;
  __syncthreads();
  if (t == 0) {
    int acc = 0;
    for (int i = 0; i < 64; ++i) { pre[i] = acc; acc += cnt[i]; }
    pre[64] = acc;
  }
  __syncthreads();
  const int total = pre[64] < 512 ? pre[64] : 512;
  int pos = pre[t];
  for (int j = 0; j < 64; ++j) {
    if (labels1[j] == l1) {
      if (pos < 512) { si[pos] = t; sj[pos] = j; }
      ++pos;
    }
  }
  for (int p = total + t; p < 512; p += 64) { si[p] = 0; sj[p] = 0; }
  __syncthreads();
  for (int p = t; p < 512; p += 64) {
    const int ii = si[p], jj = sj[p];
    const int la = labels0[ii], lb = labels0[jj];
    const int lo = la < lb ? la : lb, hi = la < lb ? lb : la;
    // pair-LUT closed form: idx = 1 + lo*(2D-lo-1)/2 + (hi-lo-1), D=10
    const int cls = (la == lb) ? 0 : (1 + lo * (19 - lo) / 2 + (hi - lo - 1));
    const int v = (p < total) ? 1 : 0;
    g_i[p] = ii; g_j[p] = jj; g_y[p] = cls; g_valid[p] = v;
    atomicAdd(&cb_s[jj], 1);              // multiplicity over ALL 512 m's
    if (v) atomicAdd(&cc_s[cls], 1.0f);   // class counts over valid pairs
  }
  __syncthreads();
  g_cb[t] = (float)cb_s[t];
  if (t < 46) g_cls[t] = cc_s[t];
}

// ---------------------------------------------------------------------------
// Kernel 2: QKV projection on the 64 distinct samples.
// out[s,n] = sum_e F[e,s] * W[n,e] + b[n], n in [0,768): q|k|v x h x d.
// Grid: 64 blocks (one per sample), 256 threads = 8 waves.
// Each wave: 6 N-tiles x 2 M-tiles, K=256 via 64 x wmma_f32_16x16x4_f32,
// software-pipelined: next-step A (LDS) + B (global) fragments are fetched
// while the current step's 12 WMMAs execute.
// ---------------------------------------------------------------------------
__global__ void __launch_bounds__(256, 1)
PSL_qkv_kernel(const float* __restrict__ feat,  // (64,256,32)
               const float* __restrict__ W,     // (768,256)
               const float* __restrict__ bias,  // (768,)
               float* __restrict__ Qo, float* __restrict__ Ko,
               float* __restrict__ Vo) {
  __shared__ float F[256 * 32];  // [e][s], 32 KB, matches global layout
  const int bi = blockIdx.x;
  const int tid = threadIdx.x;
  {
    const float4* src = (const float4*)(feat + bi * 8192);
    float4* dst = (float4*)F;
    for (int k = tid; k < 2048; k += 256) dst[k] = src[k];
  }
  __syncthreads();
  const int wave = tid >> 5, lane = tid & 31;
  const int lhalf = lane >> 4, lx = lane & 15;

  // per-lane W row base (row n, starting at k-offset 2*lhalf)
  const float* wbase[6];
#pragma unroll
  for (int q = 0; q < 6; ++q)
    wbase[q] = W + ((wave * 6 + q) * 16 + lx) * 256 + 2 * lhalf;

  v8f acc[2][6];
#pragma unroll
  for (int m = 0; m < 2; ++m)
#pragma unroll
    for (int q = 0; q < 6; ++q) acc[m][q] = vz8();

  v2f a0n, a1n, bn[6];  // "next" fragments
  auto loadfrags = [&](int k0, v2f& x0, v2f& x1, v2f* bf) {
    const int ka = k0 + 2 * lhalf;
    x0[0] = F[ka * 32 + lx];
    x0[1] = F[(ka + 1) * 32 + lx];
    x1[0] = F[ka * 32 + 16 + lx];
    x1[1] = F[(ka + 1) * 32 + 16 + lx];
#pragma unroll
    for (int q = 0; q < 6; ++q) {
      bf[q][0] = wbase[q][k0];
      bf[q][1] = wbase[q][k0 + 1];
    }
  };
  auto domma = [&](v2f a0, v2f a1, const v2f* bf) {
#pragma unroll
    for (int q = 0; q < 6; ++q) {
      acc[0][q] = wmma4(a0, bf[q], acc[0][q]);
      acc[1][q] = wmma4(a1, bf[q], acc[1][q]);
    }
  };

  loadfrags(0, a0n, a1n, bn);
  for (int k0 = 0; k0 < 252; k0 += 4) {
    v2f a0 = a0n, a1 = a1n, bc[6];
#pragma unroll
    for (int q = 0; q < 6; ++q) bc[q] = bn[q];
    loadfrags(k0 + 4, a0n, a1n, bn);  // prefetch next step
    domma(a0, a1, bc);                // compute current step
  }
  domma(a0n, a1n, bn);  // epilogue (k0 = 252)

  // write D tiles + bias into [s][h][a][d] layout
#pragma unroll
  for (int q = 0; q < 6; ++q) {
    const int n0 = (wave * 6 + q) * 16;
    const int n = n0 + lx;
    const float bn2 = bias[n];
    const int buf = n0 >> 8;          // 0=q, 1=k, 2=v
    const int h = (n0 >> 4) & 15;
    float* outp = (buf == 0) ? Qo : ((buf == 1) ? Ko : Vo);
#pragma unroll
    for (int m = 0; m < 2; ++m)
#pragma unroll
      for (int r = 0; r < 8; ++r) {
        const int s = m * 16 + r + 8 * lhalf;
        outp[((s * 16 + h) * 64 + bi) * 16 + lx] = acc[m][q][r] + bn2;
      }
  }
}

// ---------------------------------------------------------------------------
// Kernel 3: per-(s,h) attention over 64 distinct samples with key
// multiplicities c_b.  Grid: 512 blocks = (s,h), 128 threads = 4 waves.
// G = (Q/4) K^T (64x64, K=16), E = c_b*exp(G-max), O = E*V / rowsum.
// ---------------------------------------------------------------------------
__global__ void __launch_bounds__(128, 1)
PSL_attn_kernel(const float* __restrict__ Qg,
                const float* __restrict__ Kg,
                const float* __restrict__ Vg,
                const float* __restrict__ cb,
                float* __restrict__ O) {
  __shared__ float sQ[1024], sK[1024], sV[1024];
  __shared__ float sE[4096];
  __shared__ float rsum[64];
  __shared__ float cbs[64];
  const int sh = blockIdx.x;          // s*16 + h
  const int s = sh >> 4, h = sh & 15;
  const int tid = threadIdx.x;        // 128
  {
    const float4* q4 = (const float4*)(Qg + sh * 1024);
    const float4* k4 = (const float4*)(Kg + sh * 1024);
    const float4* v4 = (const float4*)(Vg + sh * 1024);
    for (int k = tid; k < 256; k += 128) {
      float4 q = q4[k];
      q.x *= 0.25f; q.y *= 0.25f; q.z *= 0.25f; q.w *= 0.25f;  // 1/sqrt(hd)
      ((float4*)sQ)[k] = q;
      ((float4*)sK)[k] = k4[k];
      ((float4*)sV)[k] = v4[k];
    }
    if (tid < 64) cbs[tid] = cb[tid];
  }
  __syncthreads();
  const int wave = tid >> 5, lane = tid & 31;
  const int lhalf = lane >> 4, lx = lane & 15;

  // G tiles: wave owns M-rows [16w,16w+16), all 4 N-tiles.
  v8f g[4];
  g[0] = vz8(); g[1] = vz8(); g[2] = vz8(); g[3] = vz8();
#pragma unroll
  for (int k0 = 0; k0 < 16; k0 += 4) {
    const int ka = k0 + 2 * lhalf;
    v2f a;
    a[0] = sQ[(wave * 16 + lx) * 16 + ka];
    a[1] = sQ[(wave * 16 + lx) * 16 + ka + 1];
#pragma unroll
    for (int nt = 0; nt < 4; ++nt) {
      v2f b;
      b[0] = sK[(nt * 16 + lx) * 16 + ka];
      b[1] = sK[(nt * 16 + lx) * 16 + ka + 1];
      g[nt] = wmma4(a, b, g[nt]);
    }
  }
  // weighted softmax numerators + row sums (rows owned entirely by this wave)
#pragma unroll
  for (int r = 0; r < 8; ++r) {
    const int row = wave * 16 + r + 8 * lhalf;
    float mr = fmaxf(fmaxf(g[0][r], g[1][r]), fmaxf(g[2][r], g[3][r]));
    mr = fmaxf(mr, __shfl_xor(mr, 1, 32));
    mr = fmaxf(mr, __shfl_xor(mr, 2, 32));
    mr = fmaxf(mr, __shfl_xor(mr, 4, 32));
    mr = fmaxf(mr, __shfl_xor(mr, 8, 32));
    float sr = 0.f;
#pragma unroll
    for (int nt = 0; nt < 4; ++nt) {
      const float ev = cbs[nt * 16 + lx] * __expf(g[nt][r] - mr);
      sE[row * 64 + nt * 16 + lx] = ev;
      sr += ev;
    }
    sr += __shfl_xor(sr, 1, 32);
    sr += __shfl_xor(sr, 2, 32);
    sr += __shfl_xor(sr, 4, 32);
    sr += __shfl_xor(sr, 8, 32);
    if (lx == 0) rsum[row] = sr;
  }
  __syncthreads();
  // O tile = E * V (M=64, N=16, K=64), then divide by rowsum
  v8f o = vz8();
  for (int k0 = 0; k0 < 64; k0 += 4) {
    const int ka = k0 + 2 * lhalf;
    v2f a;
    a[0] = sE[(wave * 16 + lx) * 64 + ka];
    a[1] = sE[(wave * 16 + lx) * 64 + ka + 1];
    v2f b;
    b[0] = sV[ka * 16 + lx];
    b[1] = sV[(ka + 1) * 16 + lx];
    o = wmma4(a, b, o);
  }
#pragma unroll
  for (int r = 0; r < 8; ++r) {
    const int row = wave * 16 + r + 8 * lhalf;
    O[(row * 32 + s) * 256 + h * 16 + lx] = o[r] / rsum[row];
  }
}

// ---------------------------------------------------------------------------
// Kernel 4: fused final-reduce + out_proj + header + weighted-CE loss.
// h[a,e] = sum_e' (sum_s fw[s]*O[a,s,e']) * Wout[e,e'] + sum(fw)*bout[e] + fb
// logits = h * Hw^T + hb ; loss over 512 pairs.  1 block x 256 threads.
// ---------------------------------------------------------------------------
__global__ void __launch_bounds__(256, 1)
PSL_head_loss_kernel(
    const float* __restrict__ O, const float* __restrict__ Wout,
    const float* __restrict__ bout, const float* __restrict__ fw,
    const float* __restrict__ fb, const float* __restrict__ Hw,
    const float* __restrict__ hb, const int* __restrict__ i_idx,
    const int* __restrict__ y, const int* __restrict__ valid,
    const float* __restrict__ cls_cnt, float* __restrict__ out) {
  extern __shared__ float sm[];
  float* Ofin = sm;            // 64*256
  float* hbuf = sm + 16384;    // 64*256
  float* logit = sm + 32768;   // 64*48
  float* rmx = sm + 35840;     // 64
  float* rls = sm + 35904;     // 64
  float* redc = sm + 35968;    // 2
  const int tid = threadIdx.x;
  const int wave = tid >> 5, lane = tid & 31;
  const int lhalf = lane >> 4, lx = lane & 15;

  float fwl[32];
#pragma unroll
  for (int s = 0; s < 32; ++s) fwl[s] = fw[s];
  float sumfw = 0.f;
#pragma unroll
  for (int s = 0; s < 32; ++s) sumfw += fwl[s];
  const float fbv = fb[0];

  // phase 1: Ofin[a][e'] = sum_s fw[s]*O[a][s][e']  (coalesced over e'=tid)
  for (int a = 0; a < 64; ++a) {
    float acc = 0.f;
#pragma unroll
    for (int s = 0; s < 32; ++s) acc += fwl[s] * O[(a * 32 + s) * 256 + tid];
    Ofin[a * 256 + tid] = acc;
  }
  if (tid == 0) { redc[0] = 0.f; redc[1] = 0.f; }
  __syncthreads();

  // phase 2: h = Ofin * Wout^T + bias  (M=64, N=256, K=256)
  {
    v8f acc[4][2];
#pragma unroll
    for (int mt = 0; mt < 4; ++mt)
#pragma unroll
      for (int q = 0; q < 2; ++q) acc[mt][q] = vz8();
    for (int k0 = 0; k0 < 256; k0 += 4) {
      const int ka = k0 + 2 * lhalf;
      v2f a[4];
#pragma unroll
      for (int mt = 0; mt < 4; ++mt) {
        a[mt][0] = Ofin[(mt * 16 + lx) * 256 + ka];
        a[mt][1] = Ofin[(mt * 16 + lx) * 256 + ka + 1];
      }
#pragma unroll
      for (int q = 0; q < 2; ++q) {
        const int n = (wave * 2 + q) * 16 + lx;
        v2f b; b[0] = Wout[n * 256 + ka]; b[1] = Wout[n * 256 + ka + 1];
#pragma unroll
        for (int mt = 0; mt < 4; ++mt) acc[mt][q] = wmma4(a[mt], b, acc[mt][q]);
      }
    }
#pragma unroll
    for (int q = 0; q < 2; ++q) {
      const int n = (wave * 2 + q) * 16 + lx;
      const float bn = bout[n] * sumfw + fbv;
#pragma unroll
      for (int mt = 0; mt < 4; ++mt)
#pragma unroll
        for (int r = 0; r < 8; ++r)
          hbuf[(mt * 16 + r + 8 * lhalf) * 256 + n] = acc[mt][q][r] + bn;
    }
  }
  __syncthreads();

  // phase 3: logits = h * Hw^T + hb (M=64, N=48 padded, K=256).
  // All 8 waves run uniformly; waves w and w+4 compute identical tiles
  // (duplicate writes of identical values) to keep EXEC all-ones for WMMA.
  {
    const int mt = wave & 3;
    v8f acc[3];
#pragma unroll
    for (int q = 0; q < 3; ++q) acc[q] = vz8();
    for (int k0 = 0; k0 < 256; k0 += 4) {
      const int ka = k0 + 2 * lhalf;
      v2f a;
      a[0] = hbuf[(mt * 16 + lx) * 256 + ka];
      a[1] = hbuf[(mt * 16 + lx) * 256 + ka + 1];
#pragma unroll
      for (int q = 0; q < 3; ++q) {
        const int n = q * 16 + lx;
        v2f b;
        b[0] = (n < 46) ? Hw[n * 256 + ka] : 0.f;
        b[1] = (n < 46) ? Hw[n * 256 + ka + 1] : 0.f;
        acc[q] = wmma4(a, b, acc[q]);
      }
    }
#pragma unroll
    for (int q = 0; q < 3; ++q) {
      const int n = q * 16 + lx;
      const float bn = (n < 46) ? hb[n] : 0.f;
#pragma unroll
      for (int r = 0; r < 8; ++r)
        logit[(mt * 16 + r + 8 * lhalf) * 48 + n] = acc[q][r] + bn;
    }
  }
  __syncthreads();

  // phase 4: per-row log-softmax stats over 46 classes
  if (tid < 64) {
    float mx = -3.4e38f;
    for (int k = 0; k < 46; ++k) mx = fmaxf(mx, logit[tid * 48 + k]);
    float sum = 0.f;
    for (int k = 0; k < 46; ++k) sum += __expf(logit[tid * 48 + k] - mx);
    rmx[tid] = mx;
    rls[tid] = logf(sum);
  }
  __syncthreads();

  // phase 5: weighted NLL over 512 pairs
  for (int p = tid; p < 512; p += 256) {
    if (valid[p]) {
      const int a = i_idx[p], yv = y[p];
      const float nll = -(logit[a * 48 + yv] - rmx[a] - rls[a]);
      const float wv = 1.0f / cls_cnt[yv];
      atomicAdd(&redc[0], wv * nll);
      atomicAdd(&redc[1], wv);
    }
  }
  __syncthreads();
  if (tid == 0) out[0] = redc[0] / redc[1];
}

// ---------------------------------------------------------------------------
extern "C" void kernel_launch(void* const* d_in, const int* in_sizes, int n_in,
                              void* d_out, int out_size, void* d_ws,
                              size_t ws_size, hipStream_t stream) {
  (void)in_sizes; (void)n_in; (void)out_size; (void)ws_size;
  const float* feat    = (const float*)d_in[0];
  const int*   labels0 = (const int*)d_in[1];
  const int*   labels1 = (const int*)d_in[2];
  const float* in_w    = (const float*)d_in[3];
  const float* in_b    = (const float*)d_in[4];
  const float* out_w   = (const float*)d_in[5];
  const float* out_b   = (const float*)d_in[6];
  const float* fin_w   = (const float*)d_in[7];
  const float* fin_b   = (const float*)d_in[8];
  const float* hdr_w   = (const float*)d_in[9];
  const float* hdr_b   = (const float*)d_in[10];

  float* ws = (float*)d_ws;  // needs ~8.41 MB
  int*   g_i     = (int*)(ws + 0);
  int*   g_j     = (int*)(ws + 512);
  int*   g_y     = (int*)(ws + 1024);
  int*   g_valid = (int*)(ws + 1536);
  float* g_cb    = ws + 2048;
  float* g_cls   = ws + 2112;
  float* Qb      = ws + 4096;
  float* Kb      = Qb + 524288;
  float* Vb      = Kb + 524288;
  float* Ob      = Vb + 524288;

  PSL_pairs_kernel<<<1, 64, 0, stream>>>(labels0, labels1, g_i, g_j, g_y,
                                         g_valid, g_cb, g_cls);
  PSL_qkv_kernel<<<64, 256, 0, stream>>>(feat, in_w, in_b, Qb, Kb, Vb);
  PSL_attn_kernel<<<512, 128, 0, stream>>>(Qb, Kb, Vb, g_cb, Ob);
  const size_t smem = 35970u * sizeof(float);  // ~140.5 KB dynamic LDS
  PSL_head_loss_kernel<<<1, 256, smem, stream>>>(Ob, out_w, out_b, fin_w,
                                                 fin_b, hdr_w, hdr_b, g_i,
                                                 g_y, g_valid, g_cls,
                                                 (float*)d_out);
}